// ArgmaxDeduplicateSlateSampler_10256381903366
// MI455X (gfx1250) — compile-verified
//
#include <hip/hip_runtime.h>

// ArgmaxDeduplicateSlateSampler for MI455X (gfx1250).
//
// Phase 1: each (b,j) vector split into 2 halves; one wave32 per half streams
//   its 200 KB through a 24-deep global_load_async_to_lds_b128 pipeline and
//   maintains a wave-wide sorted top-32 (lane k = k-th best, value desc /
//   index asc). Per element: 1 compare vs a uniform (readlane) threshold;
//   inserts (expected ~267 per 50K elements) are O(1) lane-shifts.
// Phase 2: per batch row, 20 sequential dedup picks via 2-pointer merge of
//   the two sorted 32-lists. Exact: at step j only j<=19 items are masked, so
//   the masked argmax is within the global top-21 (subset of the lists).
//
// Roofline: 512 MB single-pass / 23.3 TB/s ~= 22 us. Wave-pooled selection
// keeps VALU work (~12K cyc/wave, 5 waves/SIMD) at parity with memory time.

#define B_ 64
#define S_ 20
#define V_ 100000
#define HALF_N 50000                    // V_/2, per-wave pool
#define HALF_BYTES (HALF_N * 4)
#define THREADS 32                      // one wave per block
#define DEPTH 24                        // async tiles in flight (ASYNCcnt <= 63)
#define TILE_FLOATS 128                 // 32 lanes * 4 floats (b128 per lane)
#define TILE_BYTES 512
#define NUM_TILES ((HALF_N + TILE_FLOATS - 1) / TILE_FLOATS)   // 391

#define STR_(x) #x
#define STR(x) STR_(x)
#define DEPTH_M1_STR STR(23)

// Lexicographic "a strictly better than b": value desc, index asc on ties
// (matches jnp.argmax first-occurrence semantics).
__device__ __forceinline__ bool lex_gt(float av, int ai, float bv, int bi) {
  return (av > bv) || (av == bv && ai < bi);
}

__device__ __forceinline__ float rdlane_f(float v, int lane) {
  return __int_as_float(__builtin_amdgcn_readlane(__float_as_int(v), lane));
}

__global__ __launch_bounds__(THREADS)
void ArgmaxDedup_topk_phase1(const float* __restrict__ probs,
                             float* __restrict__ cand_val,
                             int*   __restrict__ cand_idx) {
  __shared__ __align__(16) float buf[DEPTH * TILE_FLOATS];   // 12 KB / wave

  const int lane = threadIdx.x;                  // 0..31, single wave
  const int gb   = blockIdx.x;                   // vec*2 + half
  const int vec  = gb >> 1;
  const int half = gb & 1;

  const unsigned long long base = (unsigned long long)(const void*)
      (probs + (size_t)vec * V_ + (size_t)half * HALF_N);
  const unsigned lane_byte = (unsigned)lane * 16u;
  const unsigned lds_base  = (unsigned)(unsigned long long)(const void*)&buf[0];
  const unsigned maxOff    = (unsigned)(HALF_BYTES - 16);

  // Wave-wide sorted top-32: lane k holds k-th best (lv,li).
  float lv = -3.4e38f;
  int   li = 0x7fffffff;
  float thr_v = -3.4e38f;                        // uniform copy of slot 31
  int   thr_i = 0x7fffffff;

  // ---- streaming pipeline -------------------------------------------------
  // prologue: fill DEPTH slots
#pragma unroll 1
  for (int t = 0; t < DEPTH; ++t) {
    unsigned goff = (unsigned)t * TILE_BYTES + lane_byte;
    if (goff > maxOff) goff = maxOff;
    unsigned laddr = lds_base + (unsigned)t * TILE_BYTES + lane_byte;
    asm volatile("global_load_async_to_lds_b128 %0, %1, %2"
                 :: "v"(laddr), "v"(goff), "s"(base) : "memory");
  }

  int slot = 0;                                  // rotating consume/reissue slot
#pragma unroll 1
  for (int t = 0; t < NUM_TILES; ++t) {
    // Retire oldest tile (in-order ASYNCcnt), then read it from LDS.
    if (t < NUM_TILES - DEPTH) {
      asm volatile("s_wait_asynccnt " DEPTH_M1_STR ::: "memory");
    } else if (t == NUM_TILES - DEPTH) {
      asm volatile("s_wait_asynccnt 0" ::: "memory");
    }
    const float4 q = *reinterpret_cast<const float4*>(&buf[slot * TILE_FLOATS + lane * 4]);

    // Reissue into the slot we just read (q is in VGPRs; guard LDS read done).
    if (t + DEPTH < NUM_TILES) {
      int tn = t + DEPTH;
      unsigned goff = (unsigned)tn * TILE_BYTES + lane_byte;
      if (goff > maxOff) goff = maxOff;
      unsigned laddr = lds_base + (unsigned)slot * TILE_BYTES + lane_byte;
      asm volatile("s_wait_dscnt 0\n\t"
                   "global_load_async_to_lds_b128 %0, %1, %2"
                   :: "v"(laddr), "v"(goff), "s"(base) : "memory");
    }
    slot = (slot + 1 == DEPTH) ? 0 : slot + 1;

    // Fast skip: does any lane hold a possible contender? (>= covers ties)
    float m4 = fmaxf(fmaxf(q.x, q.y), fmaxf(q.z, q.w));
    if (__ballot(m4 >= thr_v) == 0ull) continue;

    const int ebase = t * TILE_FLOATS + lane * 4; // half-relative index
#pragma unroll
    for (int e = 0; e < 4; ++e) {
      float ev = (e == 0) ? q.x : (e == 1) ? q.y : (e == 2) ? q.z : q.w;
      int   ei = ebase + e;
      bool pass = (ei < HALF_N) && lex_gt(ev, ei, thr_v, thr_i);
      unsigned mask = (unsigned)__ballot(pass);
      while (mask) {
        int src = __builtin_ctz(mask);           // uniform
        float cv = rdlane_f(ev, src);            // uniform candidate
        int   ci = __builtin_amdgcn_readlane(ei, src);
        // O(1) wave insert into sorted list: shift-down from insertion point.
        float upv = __shfl_up(lv, 1, 32);
        int   upi = __shfl_up(li, 1, 32);
        if (lane == 0) { upv = __builtin_inff(); upi = -1; }
        bool keep = lex_gt(lv, li, cv, ci);      // slot still better than cand
        bool take = lex_gt(upv, upi, cv, ci);    // cand lands exactly here
        lv = keep ? lv : (take ? cv : upv);
        li = keep ? li : (take ? ci : upi);
        thr_v = rdlane_f(lv, 31);                // new 32nd-best threshold
        thr_i = __builtin_amdgcn_readlane(li, 31);
        pass = pass && (lane != src) && lex_gt(ev, ei, thr_v, thr_i);
        mask = (unsigned)__ballot(pass);
      }
    }
  }

  // Sorted list is already laid out across lanes: store slot directly.
  cand_val[gb * 32 + lane] = lv;
  cand_idx[gb * 32 + lane] = (li == 0x7fffffff) ? li : (li + half * HALF_N);
}

// Phase 2: per batch row, 20 sequential picks via 2-pointer merge of the two
// sorted 32-lists, skipping already-selected indices.
__global__ __launch_bounds__(64)
void ArgmaxDedup_slate_phase2(const float* __restrict__ cand_val,
                              const int* __restrict__ cand_idx,
                              int* __restrict__ out) {
  const int b = threadIdx.x;
  if (b >= B_) return;
  int sel[S_];
#pragma unroll 1
  for (int j = 0; j < S_; ++j) {
    const int vec = b * S_ + j;
    const float* Av = cand_val + (vec * 2 + 0) * 32;
    const int*   Ai = cand_idx + (vec * 2 + 0) * 32;
    const float* Bv = cand_val + (vec * 2 + 1) * 32;
    const int*   Bi = cand_idx + (vec * 2 + 1) * 32;
    int pA = 0, pB = 0, chosen = -1;
#pragma unroll 1
    for (int it = 0; it < 64 && chosen < 0; ++it) {
      float av = Av[pA < 32 ? pA : 31]; int ai = Ai[pA < 32 ? pA : 31];
      float bv = Bv[pB < 32 ? pB : 31]; int bi = Bi[pB < 32 ? pB : 31];
      bool takeA = (pB >= 32) || ((pA < 32) && lex_gt(av, ai, bv, bi));
      int cand = takeA ? ai : bi;
      if (takeA) ++pA; else ++pB;
      bool used = false;
#pragma unroll 1
      for (int k = 0; k < j; ++k) used = used || (sel[k] == cand);
      if (!used) chosen = cand;
    }
    sel[j] = chosen;
    out[b * S_ + j] = chosen;                    // int32 indices (reference dtype)
  }
}

extern "C" void kernel_launch(void* const* d_in, const int* in_sizes, int n_in,
                              void* d_out, int out_size, void* d_ws, size_t ws_size,
                              hipStream_t stream) {
  (void)in_sizes; (void)n_in; (void)out_size; (void)ws_size;
  const float* probs = (const float*)d_in[0];          // [B,S,V] f32
  int* out = (int*)d_out;                              // [B,S] int32 indices

  // Workspace: 2560 half-pools * 32 entries: values then indices (~655 KB).
  const size_t npool = (size_t)B_ * S_ * 2 * 32;
  float* cval = (float*)d_ws;
  int*   cidx = (int*)((char*)d_ws + npool * sizeof(float));

  ArgmaxDedup_topk_phase1<<<B_ * S_ * 2, THREADS, 0, stream>>>(probs, cval, cidx);
  ArgmaxDedup_slate_phase2<<<1, 64, 0, stream>>>(cval, cidx, out);
}